// GridSamplingOp_79310866088165
// MI455X (gfx1250) — compile-verified
//
#include <hip/hip_runtime.h>
#include <cstdint>

// Problem constants (from reference setup_inputs)
#define BATCH 8
#define CHAN  16
#define H     512
#define W     512
#define HG    512
#define WG    512
#define SPATIAL (HG * WG)            // 262144 = 2^18 sites per batch
#define BLOCK 256                     // 8 wave32 waves per block
#define BLOCKS_PER_BATCH (SPATIAL / BLOCK)  // 1024

__global__ __launch_bounds__(BLOCK)
void grid_sample_bilinear_kernel(const float* __restrict__ x,
                                 const float* __restrict__ grid,
                                 float* __restrict__ out)
{
    __shared__ float sgrid[BLOCK * 2];   // 2 KB staging of this block's grid coords

    const int tid = threadIdx.x;

    // batch index is uniform per block -> SGPR base addresses for gathers/stores
    const int b = (int)(blockIdx.x >> 10);                 // / BLOCKS_PER_BATCH
    const int s = (int)((blockIdx.x & 1023) << 8) + tid;   // spatial site in batch

    // ---- CDNA5 async path: stage grid (float2 per lane) into LDS ----
    const uint64_t gaddr =
        (uint64_t)(uintptr_t)(grid + ((size_t)blockIdx.x * BLOCK + tid) * 2);
    const uint32_t laddr = (uint32_t)(uintptr_t)(&sgrid[tid * 2]); // LDS byte offset
    asm volatile("global_load_async_to_lds_b64 %0, %1, off"
                 :: "v"(laddr), "v"(gaddr)
                 : "memory");
    asm volatile("s_wait_asynccnt 0x0" ::: "memory");

    const float gx = sgrid[tid * 2 + 0];
    const float gy = sgrid[tid * 2 + 1];

    // ---- unnormalize (align_corners=False): ix = ((gx+1)*W - 1)/2 ----
    const float ix = (gx + 1.0f) * (0.5f * (float)W) - 0.5f;
    const float iy = (gy + 1.0f) * (0.5f * (float)H) - 0.5f;

    const float x0f = floorf(ix);
    const float y0f = floorf(iy);
    const float wx1 = ix - x0f;
    const float wy1 = iy - y0f;
    const float wx0 = 1.0f - wx1;
    const float wy0 = 1.0f - wy1;

    const int x0 = (int)x0f, y0 = (int)y0f;
    const int x1 = x0 + 1,   y1 = y0 + 1;

    const bool vx0 = (x0 >= 0) && (x0 < W);
    const bool vx1 = (x1 >= 0) && (x1 < W);
    const bool vy0 = (y0 >= 0) && (y0 < H);
    const bool vy1 = (y1 >= 0) && (y1 < H);

    // zeros padding: zero the weight of any OOB corner, clamp its address
    const float w00 = (vx0 && vy0) ? wx0 * wy0 : 0.0f;
    const float w01 = (vx1 && vy0) ? wx1 * wy0 : 0.0f;
    const float w10 = (vx0 && vy1) ? wx0 * wy1 : 0.0f;
    const float w11 = (vx1 && vy1) ? wx1 * wy1 : 0.0f;

    const int x0c = min(max(x0, 0), W - 1);
    const int x1c = min(max(x1, 0), W - 1);
    const int y0c = min(max(y0, 0), H - 1);
    const int y1c = min(max(y1, 0), H - 1);

    const int o00 = y0c * W + x0c;
    const int o01 = y0c * W + x1c;
    const int o10 = y1c * W + x0c;
    const int o11 = y1c * W + x1c;

    // SGPR-uniform batch bases; per-lane 32-bit offsets -> saddr+voffset loads
    const float* __restrict__ xb = x   + (size_t)b * CHAN * H * W;
    float* __restrict__       ob = out + (size_t)b * CHAN * SPATIAL + s;

    // Amortize coordinate math over all 16 channels: 4 gathers + 4 FMA each.
#pragma unroll
    for (int c = 0; c < CHAN; ++c) {
        const float* __restrict__ p = xb + (size_t)c * (H * W);
        float v = p[o00] * w00;
        v = fmaf(p[o01], w01, v);
        v = fmaf(p[o10], w10, v);
        v = fmaf(p[o11], w11, v);
        // streaming output: nontemporal so it doesn't evict x from L2
        __builtin_nontemporal_store(v, ob + (size_t)c * SPATIAL);
    }
}

extern "C" void kernel_launch(void* const* d_in, const int* in_sizes, int n_in,
                              void* d_out, int out_size, void* d_ws, size_t ws_size,
                              hipStream_t stream) {
    const float* x    = (const float*)d_in[0];
    const float* grid = (const float*)d_in[1];
    float* out        = (float*)d_out;

    const int total_sites = BATCH * SPATIAL;          // 2,097,152 threads
    const int nblocks     = total_sites / BLOCK;      // 8192 blocks

    hipLaunchKernelGGL(grid_sample_bilinear_kernel,
                       dim3(nblocks), dim3(BLOCK), 0, stream,
                       x, grid, out);
}